// RW_GNN_46703474377020
// MI455X (gfx1250) — compile-verified
//
#include <hip/hip_runtime.h>
#include <math.h>

#define N_NODES 50000
#define N_EDGES 500000
#define DIN     128
#define HID     128
#define KSUB    16
#define SSUB    10
#define NGRAPH  128
#define MAXSTEP 3
#define NOUT    10
#define KS      (KSUB*SSUB)        // 160
#define TRI     (SSUB*(SSUB-1)/2)  // 45

typedef float v2f __attribute__((ext_vector_type(2)));
typedef float v8f __attribute__((ext_vector_type(8)));

// Guaranteed-hardware f32 atomic add at device scope (L2 RMW, coherent across
// WGPs). Default asm scope would be WGP-level -> wrong for cross-WGP sums.
__device__ __forceinline__ void atomic_add_f32_dev(float* p, float v) {
    asm volatile("global_atomic_add_f32 %0, %1, off scope:SCOPE_DEV"
                 :: "v"(p), "v"(v) : "memory");
}

// ---------------------------------------------------------------------------
// Build symmetric hidden-graph adjacency A[K][10][10] from sigmoid(theta),
// and zero the t accumulator [MAXSTEP][G][K].
// ---------------------------------------------------------------------------
__global__ __launch_bounds__(1024) void init_kernel(const float* __restrict__ theta,
                                                    float* __restrict__ Aadj,
                                                    float* __restrict__ t) {
    int tid = threadIdx.x;
    for (int i = tid; i < KSUB * SSUB * SSUB; i += 1024) Aadj[i] = 0.f;
    for (int i = tid; i < MAXSTEP * NGRAPH * KSUB; i += 1024) t[i] = 0.f;
    __syncthreads();
    if (tid < KSUB * TRI) {
        int k = tid / TRI, idx = tid % TRI;
        int i = 0, rem = idx, row = SSUB - 1;
        while (rem >= row) { rem -= row; row--; i++; }   // triu_indices(S,1) row-major
        int j = i + 1 + rem;
        float s = 1.f / (1.f + __expf(-theta[tid]));
        Aadj[k * 100 + i * 10 + j] = s;
        Aadj[k * 100 + j * 10 + i] = s;
    }
}

// ---------------------------------------------------------------------------
// WMMA f32 GEMM:  C[M,Ncols] = act(A[M,Kd] @ Bt[Ncols,Kd]^T + bias)
// One wave per 16x16 output tile, 8 waves per block.
// A-frag (16x4 f32, 2 VGPR): lane = row (lane&15), k-pair = 2*(lane>>4).
// B-frag (4x16 f32, 2 VGPR): lane = col (lane&15), k-pair = 2*(lane>>4).
// C/D (16x16 f32, 8 VGPR):   VGPR r -> row r + 8*(lane>>4), col = lane&15.
// ---------------------------------------------------------------------------
__global__ __launch_bounds__(256) void gemm_wmma_f32(const float* __restrict__ A,
                                                     const float* __restrict__ Bt,
                                                     const float* __restrict__ bias,
                                                     float* __restrict__ C,
                                                     int M, int Ncols, int Kd,
                                                     int fuse_sigmoid) {
    int wave = threadIdx.x >> 5;
    int lane = threadIdx.x & 31;
    int tile = blockIdx.x * 8 + wave;
    int colTiles = Ncols >> 4;
    int rowTile = tile / colTiles;
    int colTile = tile - rowTile * colTiles;
    if (rowTile >= (M >> 4)) return;   // wave-uniform: EXEC all-ones inside

    int r16 = lane & 15;
    int kg  = (lane >> 4) << 1;        // 0 or 2
    const float* arow = A  + (size_t)(rowTile * 16 + r16) * Kd + kg;
    const float* brow = Bt + (size_t)(colTile * 16 + r16) * Kd + kg;

    v8f c = {};
    #pragma unroll 8
    for (int kk = 0; kk < Kd; kk += 4) {
        v2f a = *(const v2f*)(arow + kk);
        v2f b = *(const v2f*)(brow + kk);
        c = __builtin_amdgcn_wmma_f32_16x16x4_f32(false, a, false, b,
                                                  (short)0, c, false, false);
    }

    int col = colTile * 16 + r16;
    float bv = bias ? bias[col] : 0.0f;
    int rowBase = rowTile * 16 + ((lane >> 4) << 3);
    #pragma unroll
    for (int r = 0; r < 8; ++r) {
        float v = c[r] + bv;
        if (fuse_sigmoid) v = 1.0f / (1.0f + __expf(-v));
        C[(size_t)(rowBase + r) * Ncols + col] = v;
    }
}

// ---------------------------------------------------------------------------
// Walk step: yw[n,k,:] = yin[n,k,:] @ A[k]  (10x10), and zero agg.
// One thread per (n,k). A staged in LDS (6.4 KB).
// ---------------------------------------------------------------------------
__global__ __launch_bounds__(256) void walk_kernel(const float* __restrict__ yin,
                                                   const float* __restrict__ Aadj,
                                                   float* __restrict__ yw,
                                                   float* __restrict__ agg,
                                                   int NK) {
    __shared__ float As[KSUB * SSUB * SSUB];
    for (int i = threadIdx.x; i < KSUB * SSUB * SSUB; i += 256) As[i] = Aadj[i];
    __syncthreads();
    int gid = blockIdx.x * 256 + threadIdx.x;
    if (gid >= NK) return;
    int k = gid & (KSUB - 1);
    size_t base = (size_t)gid * SSUB;      // == n*160 + k*10
    float yv[SSUB];
    #pragma unroll
    for (int s = 0; s < SSUB; ++s) yv[s] = yin[base + s];
    const float* Ak = As + k * 100;
    #pragma unroll
    for (int s = 0; s < SSUB; ++s) {
        float acc = 0.f;
        #pragma unroll
        for (int sp = 0; sp < SSUB; ++sp) acc += yv[sp] * Ak[sp * 10 + s];
        yw[base + s]  = acc;
        agg[base + s] = 0.f;
    }
}

// ---------------------------------------------------------------------------
// Edge scatter: agg[dst] += yw[src]  (160 f32 per edge). One wave per edge
// iteration, grid-stride; prefetch next edge's source row (global_prefetch).
// agg/yw (64 MB combined) are L2-resident -> this is L2-atomic-bound.
// ---------------------------------------------------------------------------
__global__ __launch_bounds__(256) void edge_kernel(const int* __restrict__ src,
                                                   const int* __restrict__ dst,
                                                   const float* __restrict__ yw,
                                                   float* __restrict__ agg,
                                                   int nwaves) {
    int wid  = (blockIdx.x * blockDim.x + threadIdx.x) >> 5;
    int lane = threadIdx.x & 31;
    for (int e = wid; e < N_EDGES; e += nwaves) {
        int s = src[e], d = dst[e];
        int en = e + nwaves;
        if (en < N_EDGES) {
            // Pull next source row toward the WGP while we do atomics.
            __builtin_prefetch(yw + (size_t)src[en] * KS + lane * 5, 0, 0);
        }
        const float* ps = yw  + (size_t)s * KS;
        float*       pd = agg + (size_t)d * KS;
        #pragma unroll
        for (int j = 0; j < 5; ++j)
            atomic_add_f32_dev(pd + j * 32 + lane, ps[j * 32 + lane]);
    }
}

// ---------------------------------------------------------------------------
// y <- y0 * (y0 * agg);  t[g,k] += sum_s (y0*agg).  One thread per (n,k).
// Lanes L and L+16 of a wave handle the same k for consecutive nodes; since
// graph_indicator is sorted they nearly always share g -> merge before the
// atomic (halves atomic traffic on the 2048-entry t array).
// ---------------------------------------------------------------------------
__global__ __launch_bounds__(256) void reduce_kernel(const float* __restrict__ y0,
                                                     const float* __restrict__ agg,
                                                     float* __restrict__ y,
                                                     const int* __restrict__ gind,
                                                     float* __restrict__ t_step,
                                                     int NK) {
    int gid = blockIdx.x * 256 + threadIdx.x;
    if (gid >= NK) return;
    int lane = threadIdx.x & 31;
    int n = gid >> 4;
    int k = gid & (KSUB - 1);
    size_t base = (size_t)gid * SSUB;
    float acc = 0.f;
    #pragma unroll
    for (int s = 0; s < SSUB; ++s) {
        float y0v = y0[base + s];
        float p   = y0v * agg[base + s];
        acc += p;
        y[base + s] = y0v * p;
    }
    int g = gind[n];
    float up_acc = __shfl_down(acc, 16);
    int   up_g   = __shfl_down(g, 16);
    int   dn_g   = __shfl_up(g, 16);
    bool emit;
    if (lane < 16) {
        if (up_g == g) acc += up_acc;   // absorb partner node's sum
        emit = true;
    } else {
        emit = (dn_g != g);             // partner absorbed us unless g differs
    }
    if (emit) atomic_add_f32_dev(&t_step[g * KSUB + k], acc);
}

// ---------------------------------------------------------------------------
// Tail MLP: rw[G,48] -> fc(48->128) -> BN(eval) -> ReLU -> fc(128->10).
// One block per graph.
// ---------------------------------------------------------------------------
__global__ __launch_bounds__(128) void final_kernel(const float* __restrict__ t,
                                                    const float* __restrict__ w1,
                                                    const float* __restrict__ b1,
                                                    const float* __restrict__ bn_g,
                                                    const float* __restrict__ bn_b,
                                                    const float* __restrict__ bn_m,
                                                    const float* __restrict__ bn_v,
                                                    const float* __restrict__ w2,
                                                    const float* __restrict__ b2,
                                                    float* __restrict__ out) {
    __shared__ float rw[MAXSTEP * KSUB];
    __shared__ float hv[HID];
    int g = blockIdx.x, tid = threadIdx.x;
    if (tid < MAXSTEP * KSUB)
        rw[tid] = t[(tid >> 4) * NGRAPH * KSUB + g * KSUB + (tid & 15)];
    __syncthreads();
    float acc = b1[tid];
    #pragma unroll
    for (int i = 0; i < MAXSTEP * KSUB; ++i) acc += rw[i] * w1[tid * (MAXSTEP * KSUB) + i];
    acc = (acc - bn_m[tid]) * rsqrtf(bn_v[tid] + 1e-5f) * bn_g[tid] + bn_b[tid];
    hv[tid] = fmaxf(acc, 0.f);
    __syncthreads();
    if (tid < NOUT) {
        float o = b2[tid];
        for (int j = 0; j < HID; ++j) o += hv[j] * w2[tid * HID + j];
        out[g * NOUT + tid] = o;
    }
}

// ---------------------------------------------------------------------------
extern "C" void kernel_launch(void* const* d_in, const int* in_sizes, int n_in,
                              void* d_out, int out_size, void* d_ws, size_t ws_size,
                              hipStream_t stream) {
    const float* feats = (const float*)d_in[0];
    const int*   eidx  = (const int*)  d_in[1];   // [2,E]: src then dst
    const int*   gind  = (const int*)  d_in[2];
    const float* theta = (const float*)d_in[3];
    const float* z     = (const float*)d_in[4];   // [K*S, H] row-major
    const float* fc_w  = (const float*)d_in[5];   // [H, DIN] row-major == Bt
    const float* fc_b  = (const float*)d_in[6];
    const float* w1    = (const float*)d_in[7];
    const float* b1    = (const float*)d_in[8];
    const float* bn_g  = (const float*)d_in[9];
    const float* bn_b  = (const float*)d_in[10];
    const float* bn_m  = (const float*)d_in[11];
    const float* bn_v  = (const float*)d_in[12];
    const float* w2    = (const float*)d_in[13];
    const float* b2    = (const float*)d_in[14];
    float* out = (float*)d_out;

    float* ws   = (float*)d_ws;
    float* Aadj = ws;                                   // 1600  (pad to 2048)
    float* t    = ws + 2048;                            // 6144
    float* x    = ws + 8192;                            // N*H   = 6.40M f
    float* y0   = x  + (size_t)N_NODES * HID;           // N*160 = 8.0M f
    float* y    = y0 + (size_t)N_NODES * KS;
    float* yw   = y  + (size_t)N_NODES * KS;
    float* agg  = yw + (size_t)N_NODES * KS;

    init_kernel<<<1, 1024, 0, stream>>>(theta, Aadj, t);

    // x = sigmoid(feats @ fc_w^T + fc_b)   [50000,128]
    {
        int tiles = (N_NODES / 16) * (HID / 16);        // 25000
        gemm_wmma_f32<<<(tiles + 7) / 8, 256, 0, stream>>>(
            feats, fc_w, fc_b, x, N_NODES, HID, DIN, 1);
    }
    // y0 = x @ z^T                          [50000,160]
    {
        int tiles = (N_NODES / 16) * (KS / 16);         // 31250
        gemm_wmma_f32<<<(tiles + 7) / 8, 256, 0, stream>>>(
            x, z, nullptr, y0, N_NODES, KS, HID, 0);
    }

    const int NK = N_NODES * KSUB;                      // 800000
    const int EDGE_BLOCKS = 4096;                       // 32768 waves, ~15 edges each
    const int NWAVES = EDGE_BLOCKS * 8;
    const float* yin = y0;
    for (int step = 0; step < MAXSTEP; ++step) {
        walk_kernel<<<(NK + 255) / 256, 256, 0, stream>>>(yin, Aadj, yw, agg, NK);
        edge_kernel<<<EDGE_BLOCKS, 256, 0, stream>>>(
            eidx, eidx + N_EDGES, yw, agg, NWAVES);
        reduce_kernel<<<(NK + 255) / 256, 256, 0, stream>>>(
            y0, agg, y, gind, t + step * NGRAPH * KSUB, NK);
        yin = y;
    }

    final_kernel<<<NGRAPH, 128, 0, stream>>>(t, w1, b1, bn_g, bn_b, bn_m, bn_v,
                                             w2, b2, out);
}